// Loss_42838003810647
// MI455X (gfx1250) — compile-verified
//
#include <hip/hip_runtime.h>
#include <math.h>

typedef __attribute__((ext_vector_type(2))) float v2f;
typedef __attribute__((ext_vector_type(8))) float v8f;

#define TPB 256
#define MAXG 64
#define THRESH 0.8f

__device__ __forceinline__ float iou_xyxy(float ax1, float ay1, float ax2, float ay2, float areaA,
                                          float bx1, float by1, float bx2, float by2, float areaB) {
    float ltx = fmaxf(ax1, bx1), lty = fmaxf(ay1, by1);
    float rbx = fminf(ax2, bx2), rby = fminf(ay2, by2);
    float w = fmaxf(rbx - ltx, 0.0f), h = fmaxf(rby - lty, 0.0f);
    float inter = w * h;
    float uni = areaA + areaB - inter;
    return inter / uni;
}

// ---------------- Kernel 1: per-(item, gt) argmax over anchors ----------------
__global__ void best_anchor_kernel(const float* __restrict__ boxes,
                                   const float* __restrict__ gt,
                                   int* __restrict__ best, int A, int G) {
    const int r = blockIdx.x / G;
    const int g = blockIdx.x % G;
    const float4 gb = ((const float4*)gt)[(size_t)r * G + g];
    const float gx1 = gb.x - gb.z * 0.5f, gy1 = gb.y - gb.w * 0.5f;
    const float gx2 = gb.x + gb.z * 0.5f, gy2 = gb.y + gb.w * 0.5f;
    const float gArea = (gx2 - gx1) * (gy2 - gy1);
    const float4* b4 = ((const float4*)boxes) + (size_t)r * A;

    float bestV = -1.0f;  // matches jnp.where(valid, iou, -1.0) masking floor
    int bestI = 0;
    for (int a = threadIdx.x; a < A; a += TPB) {
        float4 bb = b4[a];
        float x1 = bb.x - bb.z * 0.5f, y1 = bb.y - bb.w * 0.5f;
        float x2 = bb.x + bb.z * 0.5f, y2 = bb.y + bb.w * 0.5f;
        float aArea = (x2 - x1) * (y2 - y1);
        float v = iou_xyxy(x1, y1, x2, y2, aArea, gx1, gy1, gx2, gy2, gArea);
        if (v > bestV) { bestV = v; bestI = a; }  // strict > keeps first max (indices ascend)
    }
    __shared__ float rv[TPB];
    __shared__ int   ri[TPB];
    rv[threadIdx.x] = bestV;
    ri[threadIdx.x] = bestI;
    __syncthreads();
    for (int s = TPB / 2; s > 0; s >>= 1) {
        if (threadIdx.x < s) {
            float ov = rv[threadIdx.x + s];
            int   oi = ri[threadIdx.x + s];
            if (ov > rv[threadIdx.x] || (ov == rv[threadIdx.x] && oi < ri[threadIdx.x])) {
                rv[threadIdx.x] = ov;
                ri[threadIdx.x] = oi;
            }
        }
        __syncthreads();
    }
    if (threadIdx.x == 0) best[blockIdx.x] = ri[0];
}

// ---------------- Kernel 2: per-anchor loss, one block = 256 anchors ----------------
__global__ void loss_kernel(const float* __restrict__ boxes,
                            const float* __restrict__ classes,
                            const float* __restrict__ gt,
                            const int* __restrict__ nobj_arr,
                            const int* __restrict__ best,
                            float* __restrict__ ws_cls,
                            float* __restrict__ ws_crd,
                            int A, int G, int blocksPerItem) {
    const int r   = blockIdx.x / blocksPerItem;
    const int ab  = blockIdx.x % blocksPerItem;
    const int tid = threadIdx.x;

    __shared__ float s_raw[MAXG][4];   // raw xywh (for smooth-L1)
    __shared__ float s_xy[MAXG][4];    // xyxy
    __shared__ float s_area[MAXG];
    __shared__ int   s_best[MAXG];
    if (tid < G) {
        float4 gb = ((const float4*)gt)[(size_t)r * G + tid];
        s_raw[tid][0] = gb.x; s_raw[tid][1] = gb.y; s_raw[tid][2] = gb.z; s_raw[tid][3] = gb.w;
        float x1 = gb.x - gb.z * 0.5f, y1 = gb.y - gb.w * 0.5f;
        float x2 = gb.x + gb.z * 0.5f, y2 = gb.y + gb.w * 0.5f;
        s_xy[tid][0] = x1; s_xy[tid][1] = y1; s_xy[tid][2] = x2; s_xy[tid][3] = y2;
        s_area[tid] = (x2 - x1) * (y2 - y1);
        s_best[tid] = best[r * G + tid];
    }
    __syncthreads();

    int nobj = nobj_arr[r];
    if (nobj > G) nobj = G;

    const int a = ab * TPB + tid;
    float4 bb = ((const float4*)boxes)[(size_t)r * A + a];
    float braw[4] = {bb.x, bb.y, bb.z, bb.w};
    float x1 = bb.x - bb.z * 0.5f, y1 = bb.y - bb.w * 0.5f;
    float x2 = bb.x + bb.z * 0.5f, y2 = bb.y + bb.w * 0.5f;
    float aArea = (x2 - x1) * (y2 - y1);

    bool anchor_pos = false;
    float crd = 0.0f;
    for (int g = 0; g < nobj; ++g) {
        float v = iou_xyxy(x1, y1, x2, y2, aArea,
                           s_xy[g][0], s_xy[g][1], s_xy[g][2], s_xy[g][3], s_area[g]);
        bool pp = (v > THRESH) || (a == s_best[g]);
        if (pp) {
            anchor_pos = true;
            float acc = 0.0f;
#pragma unroll
            for (int c = 0; c < 4; ++c) {
                float d  = braw[c] - s_raw[g][c];
                float ad = fabsf(d);
                acc += (ad < 1.0f) ? 0.5f * d * d : (ad - 0.5f);
            }
            crd += acc;
        }
    }

    float2 cl = ((const float2*)classes)[(size_t)r * A + a];
    float p  = anchor_pos ? cl.y : cl.x;
    float om = 1.0f - p;
    float cls = -(om * om) * logf(p);

    __shared__ float rc[TPB];
    __shared__ float rd[TPB];
    rc[tid] = cls;
    rd[tid] = crd;
    __syncthreads();
    for (int s = TPB / 2; s > 0; s >>= 1) {
        if (tid < s) { rc[tid] += rc[tid + s]; rd[tid] += rd[tid + s]; }
        __syncthreads();
    }
    if (tid == 0) {
        ws_cls[blockIdx.x] = rc[0];
        ws_crd[blockIdx.x] = rd[0];
    }
}

// ---------------- Kernel 2b: zero-fill pad region of partial arrays ----------------
__global__ void pad_zero_kernel(float* __restrict__ ws_cls, float* __restrict__ ws_crd,
                                int start, int end) {
    int i = start + blockIdx.x * blockDim.x + threadIdx.x;
    if (i < end) {
        ws_cls[i] = 0.0f;
        ws_crd[i] = 0.0f;
    }
}

// ---------------- Kernel 3: single-wave WMMA reduction of partials ----------------
// D = Ones(16x4) x B(4x16) + C  => every D row = column sums of B. Chaining C over
// 64-element chunks gives an exact, fixed-order f32 sum through the matrix pipe.
// nPadded is a multiple of 64 and the pad region is pre-zeroed, so loads are
// unconditional coalesced float2 (global_load_b64) — no EXEC-mask branching.
__global__ void reduce_wmma_kernel(const float* __restrict__ ws_cls,
                                   const float* __restrict__ ws_crd,
                                   float* __restrict__ out, int nPadded) {
    const int lane = threadIdx.x;  // one wave32, EXEC all ones
    v2f ones; ones[0] = 1.0f; ones[1] = 1.0f;
    v8f cc = {}; v8f cd = {};
    const float2* pc = (const float2*)ws_cls;
    const float2* pd = (const float2*)ws_crd;
    for (int base = 0; base < nPadded; base += 64) {
        float2 lc = pc[(base >> 1) + lane];
        float2 ld = pd[(base >> 1) + lane];
        v2f bc, bd;
        bc[0] = lc.x; bc[1] = lc.y;
        bd[0] = ld.x; bd[1] = ld.y;
        cc = __builtin_amdgcn_wmma_f32_16x16x4_f32(false, ones, false, bc, (short)0, cc, false, false);
        cd = __builtin_amdgcn_wmma_f32_16x16x4_f32(false, ones, false, bd, (short)0, cd, false, false);
    }
    // Row 0 of D lives in VGPR0: lanes 0..15 hold the 16 column sums.
    float c0 = cc[0];
    float d0 = cd[0];
    for (int m = 8; m >= 1; m >>= 1) {
        c0 += __shfl_xor(c0, m, 32);
        d0 += __shfl_xor(d0, m, 32);
    }
    if (lane == 0) {
        out[0] = c0 + d0;  // total
        out[1] = c0;       // class loss
        out[2] = d0;       // coord loss
    }
}

extern "C" void kernel_launch(void* const* d_in, const int* in_sizes, int n_in,
                              void* d_out, int out_size, void* d_ws, size_t ws_size,
                              hipStream_t stream) {
    // inputs: [0]=threshhold (unused), [1]=batch_boxes f32 [R,A,4],
    //         [2]=batch_classes f32 [R,A,2], [3]=batch_gt f32 [R,G,4],
    //         [4]=batch_num_objects i32 [R]
    const float* boxes   = (const float*)d_in[1];
    const float* classes = (const float*)d_in[2];
    const float* gt      = (const float*)d_in[3];
    const int*   nobj    = (const int*)d_in[4];

    const int R = in_sizes[4];
    const int A = in_sizes[1] / (R * 4);
    const int G = in_sizes[3] / (R * 4);

    const int blocksPerItem = (A + TPB - 1) / TPB;      // 256 for A=65536
    const int nBlocks2      = R * blocksPerItem;        // 2048 partials
    const int nPadded       = (nBlocks2 + 63) & ~63;    // multiple of 64 for the WMMA reducer

    float* ws_cls = (float*)d_ws;
    float* ws_crd = ws_cls + nPadded;
    int*   best   = (int*)(ws_crd + nPadded);

    best_anchor_kernel<<<R * G, TPB, 0, stream>>>(boxes, gt, best, A, G);
    loss_kernel<<<nBlocks2, TPB, 0, stream>>>(boxes, classes, gt, nobj, best,
                                              ws_cls, ws_crd, A, G, blocksPerItem);
    if (nPadded > nBlocks2) {
        int padCount = nPadded - nBlocks2;
        pad_zero_kernel<<<(padCount + 63) / 64, 64, 0, stream>>>(ws_cls, ws_crd,
                                                                 nBlocks2, nPadded);
    }
    reduce_wmma_kernel<<<1, 32, 0, stream>>>(ws_cls, ws_crd, (float*)d_out, nPadded);
}